// TropicalScheduling_23295902613794
// MI455X (gfx1250) — compile-verified
//
#include <hip/hip_runtime.h>
#include <math.h>
#include <stdint.h>

// Problem constants (match reference)
#define B_SZ     256
#define IN_DIM   64
#define H        768
#define NLAYERS  4
#define P        8
#define LN_EPS   1e-5f

#define NTHREADS 256             // 8 wave32s
#define HS       (H + 4)         // padded LDS stride for hn tiles
#define OT       64              // tropical o-tile (outputs per WG)
#define WTS      (OT + 1)        // transposed W stride: 65 == 1 mod 64 banks -> conflict-free
#define ROWS_IN  16              // rows per WG, input projection
#define ROWS_T   32              // rows per WG, tropical  (halves W L2 traffic)
#define ROWS_G   32              // rows per WG, GEMM      (halves B L2 traffic)

typedef float v2f __attribute__((ext_vector_type(2)));
typedef float v8f __attribute__((ext_vector_type(8)));

#define NEG_BIG (-3.402823466e38f)
#define POS_BIG ( 3.402823466e38f)

__device__ __forceinline__ float sigmoidf_(float x) { return 1.0f / (1.0f + expf(-x)); }
__device__ __forceinline__ float geluf_(float x)    { return 0.5f * x * (1.0f + erff(x * 0.70710678118654752f)); }

// ---- CDNA5 async global->LDS copy (16B per lane), tracked on ASYNCcnt ----
// Low 32 bits of a generic pointer to LDS are
// the LDS byte address (ISA 10.2, LDS aperture).
__device__ __forceinline__ void async_copy_b128(float* dst_lds, const float* src_glb) {
    unsigned ldsa = (unsigned)(uintptr_t)dst_lds;
    asm volatile("global_load_async_to_lds_b128 %0, %1, off"
                 :: "v"(ldsa), "v"(src_glb) : "memory");
}
__device__ __forceinline__ void async_wait_all() {
    asm volatile("s_wait_asynccnt 0x0" ::: "memory");
}

// ============================================================================
// Input projection: h[B,H] = x[B,64] @ w_in[64,H] + b_in   (WMMA f32 16x16x4)
// grid: 16 WGs, 256 threads
// ============================================================================
__global__ __launch_bounds__(NTHREADS, 1)
void in_proj_kernel(const float* __restrict__ x, const float* __restrict__ w_in,
                    const float* __restrict__ b_in, float* __restrict__ h)
{
    __shared__ float s_x[ROWS_IN * (IN_DIM + 4)];
    const int tid  = threadIdx.x;
    const int lane = tid & 31;
    const int wave = tid >> 5;
    const int r0   = blockIdx.x * ROWS_IN;
    const int frow = lane & 15;
    const int fhi  = lane >> 4;
    const int kq   = fhi * 2;
    const int XS   = IN_DIM + 4;

    for (int idx = tid; idx < ROWS_IN * IN_DIM; idx += NTHREADS) {
        int r = idx >> 6, c = idx & (IN_DIM - 1);
        s_x[r * XS + c] = x[(r0 + r) * IN_DIM + c];
    }
    __syncthreads();

    for (int nt = wave * 6; nt < wave * 6 + 6; ++nt) {
        const int n = nt * 16 + frow;
        v8f acc = {};
        #pragma unroll
        for (int k0 = 0; k0 < IN_DIM; k0 += 4) {
            v2f a, b;
            a[0] = s_x[frow * XS + k0 + kq];
            a[1] = s_x[frow * XS + k0 + kq + 1];
            b[0] = w_in[(k0 + kq    ) * H + n];
            b[1] = w_in[(k0 + kq + 1) * H + n];
            acc = __builtin_amdgcn_wmma_f32_16x16x4_f32(false, a, false, b, (short)0, acc, false, false);
        }
        const float bias = b_in[n];
        #pragma unroll
        for (int r = 0; r < 8; ++r)
            h[(r0 + r + fhi * 8) * H + n] = acc[r] + bias;
    }
}

// ============================================================================
// LayerNorm (one wave per row). grid: 32 WGs -> 8 rows per WG
// ============================================================================
__global__ __launch_bounds__(NTHREADS, 1)
void ln_kernel(const float* __restrict__ h, const float* __restrict__ g,
               const float* __restrict__ b, float* __restrict__ hn)
{
    const int row  = blockIdx.x * 8 + (threadIdx.x >> 5);
    const int lane = threadIdx.x & 31;
    const float* hr = h + row * H;

    float s = 0.f, s2 = 0.f;
    for (int i = lane; i < H; i += 32) {
        float v = hr[i];
        s += v; s2 += v * v;
    }
    #pragma unroll
    for (int m = 16; m >= 1; m >>= 1) {
        s  += __shfl_xor(s,  m, 32);
        s2 += __shfl_xor(s2, m, 32);
    }
    float mu   = s  * (1.0f / H);
    float var  = s2 * (1.0f / H) - mu * mu;
    float rstd = rsqrtf(var + LN_EPS);
    for (int i = lane; i < H; i += 32)
        hn[row * H + i] = (hr[i] - mu) * rstd * g[i] + b[i];
}

// ============================================================================
// Tropical max-plus + LF dual activation.
// grid: (12 o-tiles, 8 row-tiles of 32). LDS: hn 32x768 (async-loaded) +
// W o-slice 64x768 transposed (stride 65 -> conflict-free) = 298 KB.
// 8 accumulators/thread -> W read once per 32 batch rows.
// ============================================================================
__global__ __launch_bounds__(NTHREADS, 1)
void trop_kernel(const float* __restrict__ hn, const float* __restrict__ tw,
                 const float* __restrict__ tb,
                 const float* __restrict__ am, const float* __restrict__ bm,
                 const float* __restrict__ an, const float* __restrict__ bn,
                 const float* __restrict__ alpha, float* __restrict__ tout)
{
    __shared__ float s_hn[ROWS_T * HS];    // 98.8 KB
    __shared__ float s_w [H * WTS];        // 199.7 KB (transposed: [ii][oo])
    const int tid = threadIdx.x;
    const int o0  = blockIdx.x * OT;
    const int r0  = blockIdx.y * ROWS_T;

    // hn tile: 32x768, async global->LDS (16B per lane)
    for (int idx = tid; idx < ROWS_T * (H / 4); idx += NTHREADS) {
        int r = idx / (H / 4), c4 = idx % (H / 4);
        async_copy_b128(&s_hn[r * HS + c4 * 4], &hn[(r0 + r) * H + c4 * 4]);
    }
    // W slice: 64x768, coalesced float4 loads, stored transposed
    for (int idx = tid; idx < OT * (H / 4); idx += NTHREADS) {
        int oo = idx / (H / 4), i4 = idx % (H / 4);
        float4 w = *(const float4*)&tw[(o0 + oo) * H + i4 * 4];
        s_w[(i4 * 4 + 0) * WTS + oo] = w.x;
        s_w[(i4 * 4 + 1) * WTS + oo] = w.y;
        s_w[(i4 * 4 + 2) * WTS + oo] = w.z;
        s_w[(i4 * 4 + 3) * WTS + oo] = w.w;
    }
    async_wait_all();
    __syncthreads();

    const int to  = tid & 63;   // o within tile
    const int tbq = tid >> 6;   // wave-uniform base row -> hn reads broadcast
    float acc[8];
    #pragma unroll
    for (int j = 0; j < 8; ++j) acc[j] = NEG_BIG;

    #pragma unroll 2
    for (int ii = 0; ii < H; ii += 4) {
        float w0 = s_w[(ii + 0) * WTS + to];
        float w1 = s_w[(ii + 1) * WTS + to];
        float w2 = s_w[(ii + 2) * WTS + to];
        float w3 = s_w[(ii + 3) * WTS + to];
        #pragma unroll
        for (int j = 0; j < 8; ++j) {
            float4 hv = *(const float4*)&s_hn[(tbq + 4 * j) * HS + ii];
            acc[j] = fmaxf(acc[j], hv.x + w0);
            acc[j] = fmaxf(acc[j], hv.y + w1);
            acc[j] = fmaxf(acc[j], hv.z + w2);
            acc[j] = fmaxf(acc[j], hv.w + w3);
        }
    }

    const int   o   = o0 + to;
    const float tbo = tb[o];
    const float* amp = am + o * P;
    const float* bmp = bm + o * P;
    const float* anp = an + o * P;
    const float* bnp = bn + o * P;
    const float a = sigmoidf_(alpha[o]);

    #pragma unroll
    for (int j = 0; j < 8; ++j) {
        float t  = acc[j] + tbo;
        float fx = NEG_BIG, fn = POS_BIG;
        #pragma unroll
        for (int p = 0; p < P; ++p) {
            fx = fmaxf(fx, t * amp[p] + bmp[p]);
            fn = fminf(fn, t * anp[p] + bnp[p]);
        }
        tout[(r0 + tbq + 4 * j) * H + o] = a * fx + (1.0f - a) * fn;
    }
}

// ============================================================================
// gate & cls GEMMs (WMMA f32) + fused combine, in place.
// grid: (6 n-chunks of 128, 8 row-tiles of 32). Each wave: one 16-col tile x
// two 16-row tiles -> 4 WMMAs per k-step sharing one pair of B-fragments.
// ============================================================================
__global__ __launch_bounds__(NTHREADS, 1)
void gemm_combine_kernel(const float* __restrict__ hn, float* __restrict__ h,
                         const float* __restrict__ t,
                         const float* __restrict__ gw, const float* __restrict__ gb,
                         const float* __restrict__ cw, const float* __restrict__ cb)
{
    __shared__ float s_hn[ROWS_G * HS];    // 98.8 KB
    const int tid  = threadIdx.x;
    const int lane = tid & 31;
    const int wave = tid >> 5;
    const int r0   = blockIdx.y * ROWS_G;
    const int n0   = blockIdx.x * 128 + wave * 16;
    const int frow = lane & 15;
    const int fhi  = lane >> 4;
    const int kq   = fhi * 2;

    for (int idx = tid; idx < ROWS_G * (H / 4); idx += NTHREADS) {
        int r = idx / (H / 4), c4 = idx % (H / 4);
        async_copy_b128(&s_hn[r * HS + c4 * 4], &hn[(r0 + r) * H + c4 * 4]);
    }
    async_wait_all();
    __syncthreads();

    const int n = n0 + frow;
    v8f accg0 = {}, accc0 = {}, accg1 = {}, accc1 = {};
    #pragma unroll 4
    for (int k0 = 0; k0 < H; k0 += 4) {
        v2f a0, a1, bg, bc;
        a0[0] = s_hn[frow * HS + k0 + kq];
        a0[1] = s_hn[frow * HS + k0 + kq + 1];
        a1[0] = s_hn[(16 + frow) * HS + k0 + kq];
        a1[1] = s_hn[(16 + frow) * HS + k0 + kq + 1];
        bg[0] = gw[(k0 + kq    ) * H + n];
        bg[1] = gw[(k0 + kq + 1) * H + n];
        bc[0] = cw[(k0 + kq    ) * H + n];
        bc[1] = cw[(k0 + kq + 1) * H + n];
        if (k0 + 16 < H) {
            __builtin_prefetch(&gw[(k0 + 16 + kq) * H + n], 0, 1);
            __builtin_prefetch(&cw[(k0 + 16 + kq) * H + n], 0, 1);
        }
        accg0 = __builtin_amdgcn_wmma_f32_16x16x4_f32(false, a0, false, bg, (short)0, accg0, false, false);
        accc0 = __builtin_amdgcn_wmma_f32_16x16x4_f32(false, a0, false, bc, (short)0, accc0, false, false);
        accg1 = __builtin_amdgcn_wmma_f32_16x16x4_f32(false, a1, false, bg, (short)0, accg1, false, false);
        accc1 = __builtin_amdgcn_wmma_f32_16x16x4_f32(false, a1, false, bc, (short)0, accc1, false, false);
    }
    const float gbi = gb[n], cbi = cb[n];
    #pragma unroll
    for (int mt = 0; mt < 2; ++mt) {
        v8f ag = mt ? accg1 : accg0;
        v8f ac = mt ? accc1 : accc0;
        #pragma unroll
        for (int r = 0; r < 8; ++r) {
            int   row = r0 + mt * 16 + r + fhi * 8;
            float g   = sigmoidf_(ag[r] + gbi);
            float c   = geluf_(ac[r] + cbi);
            h[row * H + n] = h[row * H + n] + g * t[row * H + n] + (1.0f - g) * c;
        }
    }
}

// ============================================================================
// Final LayerNorm + head (one wave per row). grid: 32 WGs
// ============================================================================
__global__ __launch_bounds__(NTHREADS, 1)
void head_kernel(const float* __restrict__ h, const float* __restrict__ g,
                 const float* __restrict__ b, const float* __restrict__ hw,
                 const float* __restrict__ hb, float* __restrict__ out)
{
    const int row  = blockIdx.x * 8 + (threadIdx.x >> 5);
    const int lane = threadIdx.x & 31;
    const float* hr = h + row * H;

    float s = 0.f, s2 = 0.f;
    for (int i = lane; i < H; i += 32) {
        float v = hr[i];
        s += v; s2 += v * v;
    }
    #pragma unroll
    for (int m = 16; m >= 1; m >>= 1) {
        s  += __shfl_xor(s,  m, 32);
        s2 += __shfl_xor(s2, m, 32);
    }
    float mu   = s  * (1.0f / H);
    float var  = s2 * (1.0f / H) - mu * mu;
    float rstd = rsqrtf(var + LN_EPS);

    float dot = 0.f;
    for (int i = lane; i < H; i += 32)
        dot += ((hr[i] - mu) * rstd * g[i] + b[i]) * hw[i];
    #pragma unroll
    for (int m = 16; m >= 1; m >>= 1)
        dot += __shfl_xor(dot, m, 32);
    if (lane == 0)
        out[row] = dot + hb[0];
}

// ============================================================================
extern "C" void kernel_launch(void* const* d_in, const int* in_sizes, int n_in,
                              void* d_out, int out_size, void* d_ws, size_t ws_size,
                              hipStream_t stream) {
    (void)in_sizes; (void)n_in; (void)out_size; (void)ws_size;
    const float* x        = (const float*)d_in[0];
    const float* w_in     = (const float*)d_in[1];
    const float* b_in     = (const float*)d_in[2];
    const float* ln_g     = (const float*)d_in[3];
    const float* ln_b     = (const float*)d_in[4];
    const float* trop_w   = (const float*)d_in[5];
    const float* trop_b   = (const float*)d_in[6];
    const float* lf_amax  = (const float*)d_in[7];
    const float* lf_bmax  = (const float*)d_in[8];
    const float* lf_amin  = (const float*)d_in[9];
    const float* lf_bmin  = (const float*)d_in[10];
    const float* lf_alpha = (const float*)d_in[11];
    const float* gate_w   = (const float*)d_in[12];
    const float* gate_b   = (const float*)d_in[13];
    const float* cls_w    = (const float*)d_in[14];
    const float* cls_b    = (const float*)d_in[15];
    const float* out_g    = (const float*)d_in[16];
    const float* out_b    = (const float*)d_in[17];
    const float* head_w   = (const float*)d_in[18];
    const float* head_b   = (const float*)d_in[19];
    float* out = (float*)d_out;

    // Workspace: h, hn, t  (3 x 256*768 f32 = 2.25 MB, L2-resident)
    float* h  = (float*)d_ws;
    float* hn = h  + (size_t)B_SZ * H;
    float* t  = hn + (size_t)B_SZ * H;

    in_proj_kernel<<<B_SZ / ROWS_IN, NTHREADS, 0, stream>>>(x, w_in, b_in, h);

    for (int l = 0; l < NLAYERS; ++l) {
        ln_kernel<<<B_SZ / 8, NTHREADS, 0, stream>>>(h, ln_g + l * H, ln_b + l * H, hn);

        dim3 gB(H / OT, B_SZ / ROWS_T);   // 12 x 8 = 96 WGs
        trop_kernel<<<gB, NTHREADS, 0, stream>>>(
            hn, trop_w + (size_t)l * H * H, trop_b + l * H,
            lf_amax + (size_t)l * H * P, lf_bmax + (size_t)l * H * P,
            lf_amin + (size_t)l * H * P, lf_bmin + (size_t)l * H * P,
            lf_alpha + l * H, t);

        dim3 gC(H / 128, B_SZ / ROWS_G);  // 6 x 8 = 48 WGs
        gemm_combine_kernel<<<gC, NTHREADS, 0, stream>>>(
            hn, h, t,
            gate_w + (size_t)l * H * H, gate_b + l * H,
            cls_w  + (size_t)l * H * H, cls_b  + l * H);
    }

    head_kernel<<<B_SZ / 8, NTHREADS, 0, stream>>>(h, out_g, out_b, head_w, head_b, out);
}